// ExtractSearchWindows_45930380264072
// MI455X (gfx1250) — compile-verified
//
#include <hip/hip_runtime.h>
#include <stdint.h>

// ---------------------------------------------------------------------------
// ExtractSearchWindows for MI455X (gfx1250)
//
// out[b,h,w,a*7+c,p*7+q] = (uint8) Qpad[b, h+a+p, w+c+q]
//   Qpad = zero-pad(input[:,0], 6)   (204x204 per batch, uint8 after convert)
//
// Pure data movement: 177 MB of stores, 0 FLOPs. Strategy:
//   1) pad+convert f32->u8 once into d_ws (83 KB total)
//   2) per block: TDM (tensor_load_to_lds) DMA of the 41.6 KB padded batch
//      image into LDS, s_wait_tensorcnt, barrier
//   3) grid-stride over aligned 16-byte output chunks: odometer index
//      decomposition, 16x ds_load_u8 gather, 1x non-temporal b128 store
// ---------------------------------------------------------------------------

typedef __attribute__((ext_vector_type(4))) unsigned int v4u;
typedef __attribute__((ext_vector_type(8))) int          v8i;
typedef __attribute__((ext_vector_type(4))) int          v4i;
typedef __attribute__((ext_vector_type(4))) uint32_t     uv4;

#define HH 192
#define WW 192
#define PW 204                      // padded width/height (192 + 2*6)
#define PADB (PW * PW)              // 41616 bytes: one padded batch image
#define BATCH_BYTES (36864u * 2401u) // 88,510,464 bytes of output per batch
#define CHUNKS (BATCH_BYTES / 16u)   // 5,531,904 aligned 16B chunks per batch
#define BLOCKS_PER_BATCH 256
#define THREADS 256

// --------------------------- kernel 1: pad + f32->u8 -----------------------
__global__ void pad_u8_kernel(const float* __restrict__ in,
                              unsigned char* __restrict__ q) {
    int idx = blockIdx.x * blockDim.x + threadIdx.x;
    if (idx >= 2 * PADB) return;
    int b   = idx / PADB;
    int rem = idx - b * PADB;
    int y   = rem / PW;
    int x   = rem - y * PW;
    int hy = y - 6, wx = x - 6;
    unsigned char v = 0;
    if ((unsigned)hy < (unsigned)HH && (unsigned)wx < (unsigned)WW)
        v = (unsigned char)in[(b * HH + hy) * WW + wx]; // trunc like astype(u8)
    q[idx] = v;
}

// --------------------------- kernel 2: window expansion --------------------
__global__ __launch_bounds__(THREADS)
void win_kernel(const unsigned char* __restrict__ q,
                unsigned char* __restrict__ out) {
    __shared__ unsigned char lds_q[PADB];

    const unsigned b   = blockIdx.x / BLOCKS_PER_BATCH;
    const unsigned blk = blockIdx.x % BLOCKS_PER_BATCH;

    // ---- stage padded batch image into LDS via the Tensor Data Mover ----
#if __has_builtin(__builtin_amdgcn_tensor_load_to_lds)
    if ((threadIdx.x >> 5) == 0) {   // one wave issues the DMA (EXEC ignored)
        uint64_t ga      = (uint64_t)(uintptr_t)(q + (size_t)b * PADB);
        uint32_t lds_off = (uint32_t)(uintptr_t)&lds_q[0];
        // D# group 0: count=1 | lds_addr | global_addr[56:0] | type=2
        v4u g0 = { 1u,
                   lds_off,
                   (uint32_t)(ga & 0xFFFFFFFFu),
                   (uint32_t)((ga >> 32) & 0x1FFFFFFu) | 0x80000000u };
        // D# group 1: data_size=1B, 2D tensor 204x204, tile 204x204, stride 204
        v8i g1 = { 0,                                   // mask/flags/data_size=1B
                   (int)((PW & 0xFFFFu) << 16),         // tensor_dim0[15:0]
                   (int)((PW & 0xFFFFu) << 16),         // dim0[31:16]=0 | tensor_dim1[15:0]
                   (int)((unsigned)PW << 16),           // dim1[31:16]=0 | tile_dim0
                   (int)(PW & 0xFFFFu),                 // tile_dim1 | tile_dim2=0
                   (int)PW,                             // tensor_dim0_stride[31:0]
                   0, 0 };                              // stride hi / dim1_stride
        v4i g2 = {0, 0, 0, 0};
        v4i g3 = {0, 0, 0, 0};
#if defined(__clang_major__) && (__clang_major__ >= 23)
        v8i g4 = {0, 0, 0, 0, 0, 0, 0, 0};
        __builtin_amdgcn_tensor_load_to_lds(g0, g1, g2, g3, g4, 0);
#else
        __builtin_amdgcn_tensor_load_to_lds(g0, g1, g2, g3, 0);
#endif
        __builtin_amdgcn_s_wait_tensorcnt(0);
    }
#else
    for (unsigned i = threadIdx.x; i < PADB; i += THREADS)
        lds_q[i] = q[(size_t)b * PADB + i];
#endif
    __syncthreads();

    unsigned char* outB = out + (size_t)b * BATCH_BYTES;

    // ---- each iteration: one aligned 16-byte output chunk per thread ----
    for (uint32_t ci = blk * THREADS + threadIdx.x; ci < CHUNKS;
         ci += BLOCKS_PER_BATCH * THREADS) {
        uint32_t g     = ci * 16u;
        uint32_t pixel = g / 2401u;            // h*192 + w
        uint32_t rem   = g - pixel * 2401u;    // ac*49 + pq
        uint32_t h  = pixel / 192u;
        uint32_t w  = pixel - h * 192u;
        uint32_t ac = rem / 49u;
        uint32_t pq = rem - ac * 49u;
        uint32_t a  = ac / 7u, c  = ac - a * 7u;
        uint32_t p  = pq / 7u, qq = pq - p * 7u;

        uint32_t rowbase = (h + a + p) * PW;   // LDS row byte offset
        uint32_t col     = w + c + qq;         // LDS column
        uint32_t d0 = 0, d1 = 0, d2 = 0, d3 = 0;

#pragma unroll
        for (int j = 0; j < 16; ++j) {
            uint32_t byte = (uint32_t)lds_q[rowbase + col];
            if (j < 4)       d0 |= byte << ((j & 3) * 8);
            else if (j < 8)  d1 |= byte << ((j & 3) * 8);
            else if (j < 12) d2 |= byte << ((j & 3) * 8);
            else             d3 |= byte << ((j & 3) * 8);
            // odometer advance of (q,p,c,a,w,h); col = w+c+q, rowbase=(h+a+p)*PW
            ++qq; ++col;
            if (qq == 7u) { qq = 0u; col -= 7u; ++p; rowbase += PW;
              if (p == 7u) { p = 0u; rowbase -= 7u * PW; ++c; ++col;
                if (c == 7u) { c = 0u; col -= 7u; ++a; rowbase += PW;
                  if (a == 7u) { a = 0u; rowbase -= 7u * PW; ++w; ++col;
                    if (w == 192u) { w = 0u; col -= 192u; ++h; rowbase += PW; }
                  }
                }
              }
            }
        }
        uv4 v = { d0, d1, d2, d3 };
        __builtin_nontemporal_store(v, (uv4*)(outB) + ci);
    }
}

// --------------------------- launcher --------------------------------------
extern "C" void kernel_launch(void* const* d_in, const int* in_sizes, int n_in,
                              void* d_out, int out_size, void* d_ws, size_t ws_size,
                              hipStream_t stream) {
    const float* x      = (const float*)d_in[0];  // (2,1,192,192) f32
    // d_in[1] = search_range (==3): cv=7, offset=0 baked into this kernel.
    unsigned char* qpad = (unsigned char*)d_ws;   // 2*204*204 = 83,232 B scratch
    unsigned char* out  = (unsigned char*)d_out;  // (2,192,192,49,49) u8

    int padTotal = 2 * PADB;
    pad_u8_kernel<<<(padTotal + 255) / 256, 256, 0, stream>>>(x, qpad);
    win_kernel<<<2 * BLOCKS_PER_BATCH, THREADS, 0, stream>>>(qpad, out);
}